// PrototypicalNetworkModel_41308995453753
// MI455X (gfx1250) — compile-verified
//
#include <hip/hip_runtime.h>
#include <hip/hip_bf16.h>

typedef __attribute__((ext_vector_type(2))) float v2f;
typedef __attribute__((ext_vector_type(8))) float v8f;

#define N_WAY 100
#define K_SHOT 10
#define N_SUPPORT 1000
#define N_QUERY 8192
#define F_IN 4096
#define D_EMB 1024
#define EPSV 1e-6f
#define N_PROTO_PAD 112   // 7 tiles of 16

// ---------------- workspace layout (float offsets) ----------------
#define OFF_ZQ     ((size_t)0)                         // 8192*1024
#define OFF_ZS     (OFF_ZQ + (size_t)N_QUERY * D_EMB)  // 1000*1024
#define OFF_ZT     (OFF_ZS + (size_t)N_SUPPORT * D_EMB)// 112*1024 (padded)
#define OFF_P2     (OFF_ZT + (size_t)N_PROTO_PAD * D_EMB)
#define OFF_PS     (OFF_P2 + N_PROTO_PAD)
#define OFF_Q2     (OFF_PS + N_PROTO_PAD)
#define OFF_QS     (OFF_Q2 + N_QUERY)
#define OFF_ORDER  (OFF_QS + N_QUERY)                  // 1000 ints

// ---------------- stable per-class grouping ----------------
__global__ void build_order_kernel(const int* __restrict__ labels, int* __restrict__ order) {
    int c = threadIdx.x;
    if (c >= N_WAY) return;
    int pos = 0;
    for (int i = 0; i < N_SUPPORT; ++i) {
        if (labels[i] == c) { order[c * K_SHOT + pos] = i; ++pos; }
    }
}

// ---------------- fp32 WMMA GEMM: C[M x 1024] = X[M x 4096] @ W + b --------
// block tile 128x128, K-step 32, 256 threads = 8 waves, wave tile 32x64.
// LDS layout: tile[kpair][2*m] holds (K=2*kp, K=2*kp+1) as an aligned float2,
// so each WMMA fragment is ONE ds_load_b64 into an even VGPR pair.
// Double-buffered: next tile's global loads are issued before the WMMA block.
#define PSTRIDE 288   // 2*128 + 32 pad: 288 % 64 == 32 -> half-waves in disjoint bank halves
#define KTILES_G (F_IN / 32)

__global__ __launch_bounds__(256)
void gemm_bias_kernel(const float* __restrict__ X, const float* __restrict__ Wm,
                      const float* __restrict__ bias, float* __restrict__ C, int M) {
    __shared__ float As[2][16][PSTRIDE];
    __shared__ float Bs[2][16][PSTRIDE];

    const int tid  = threadIdx.x;
    const int row0 = blockIdx.x * 128;
    const int col0 = blockIdx.y * 128;

    const int wave = tid >> 5, lane = tid & 31;
    const int lm = lane & 15, hi = lane >> 4;       // half-wave select
    const int mw = (wave & 3) * 32;                 // 4 M-groups of 32
    const int nw = (wave >> 2) * 64;                // 2 N-groups of 64

    // A-tile loader indices: 8 k-chunks of 4 per row, 128 rows
    const int arow = tid >> 3;            // 0..31
    const int akc  = (tid & 7) * 4;       // 0..28
    const int akp  = akc >> 1;            // even kpair index
    // B-tile loader indices: kpairs 0..7 (+8), 4 consecutive n
    const int bkr = tid >> 5;             // 0..7
    const int bnc = (tid & 31) * 4;       // 0..124

    float4 ra[4], rb0[2], rb1[2];         // register staging for the next tile

    auto load_global = [&](int kt) {
        const int k0 = kt * 32;
#pragma unroll
        for (int r = 0; r < 4; ++r) {
            int grow = row0 + arow + 32 * r; if (grow > M - 1) grow = M - 1;
            ra[r] = *(const float4*)(X + (size_t)grow * F_IN + k0 + akc);
        }
#pragma unroll
        for (int i = 0; i < 2; ++i) {
            const int k = k0 + 2 * (bkr + 8 * i);
            rb0[i] = *(const float4*)(Wm + (size_t)k       * D_EMB + col0 + bnc);
            rb1[i] = *(const float4*)(Wm + (size_t)(k + 1) * D_EMB + col0 + bnc);
        }
    };
    auto store_lds = [&](int buf) {
#pragma unroll
        for (int r = 0; r < 4; ++r) {
            const int row = arow + 32 * r;
            *(float2*)&As[buf][akp    ][2 * row] = make_float2(ra[r].x, ra[r].y);
            *(float2*)&As[buf][akp + 1][2 * row] = make_float2(ra[r].z, ra[r].w);
        }
#pragma unroll
        for (int i = 0; i < 2; ++i) {
            const int kp = bkr + 8 * i;
            *(float2*)&Bs[buf][kp][2 * (bnc + 0)] = make_float2(rb0[i].x, rb1[i].x);
            *(float2*)&Bs[buf][kp][2 * (bnc + 1)] = make_float2(rb0[i].y, rb1[i].y);
            *(float2*)&Bs[buf][kp][2 * (bnc + 2)] = make_float2(rb0[i].z, rb1[i].z);
            *(float2*)&Bs[buf][kp][2 * (bnc + 3)] = make_float2(rb0[i].w, rb1[i].w);
        }
    };

    v8f acc[2][4];
#pragma unroll
    for (int i = 0; i < 2; ++i)
#pragma unroll
        for (int j = 0; j < 4; ++j)
            acc[i][j] = (v8f){0.f,0.f,0.f,0.f,0.f,0.f,0.f,0.f};

    load_global(0);
    store_lds(0);
    int cur = 0;

    for (int kt = 0; kt < KTILES_G; ++kt) {
        __syncthreads();
        if (kt + 1 < KTILES_G) load_global(kt + 1);   // overlap with WMMA below

        const float (*pA)[PSTRIDE] = As[cur];
        const float (*pB)[PSTRIDE] = Bs[cur];
#pragma unroll
        for (int ks = 0; ks < 8; ++ks) {
            const int kp = ks * 2 + hi;   // lanes 0-15: K=4ks,4ks+1 ; lanes 16-31: +2,+3
            v2f a[2], b[4];
#pragma unroll
            for (int i = 0; i < 2; ++i)
                a[i] = *(const v2f*)&pA[kp][2 * (mw + 16 * i + lm)];
#pragma unroll
            for (int j = 0; j < 4; ++j)
                b[j] = *(const v2f*)&pB[kp][2 * (nw + 16 * j + lm)];
#pragma unroll
            for (int i = 0; i < 2; ++i)
#pragma unroll
                for (int j = 0; j < 4; ++j)
                    acc[i][j] = __builtin_amdgcn_wmma_f32_16x16x4_f32(
                        false, a[i], false, b[j], (short)0, acc[i][j], false, false);
        }

        if (kt + 1 < KTILES_G) store_lds(cur ^ 1);
        cur ^= 1;
    }

    // ---- epilogue: add bias, store ----
#pragma unroll
    for (int i = 0; i < 2; ++i) {
#pragma unroll
        for (int j = 0; j < 4; ++j) {
            const int col = col0 + nw + 16 * j + lm;
            const float bv = bias[col];
#pragma unroll
            for (int e = 0; e < 8; ++e) {
                const int row = row0 + mw + 16 * i + 8 * hi + e;
                if (row < M) C[(size_t)row * D_EMB + col] = acc[i][j][e] + bv;
            }
        }
    }
}

// ---------------- prototypes: median/mean over K_SHOT, class stats ----------
__global__ __launch_bounds__(256)
void proto_kernel(const float* __restrict__ zs, const int* __restrict__ order,
                  float* __restrict__ ztot, float* __restrict__ p2, float* __restrict__ ps) {
    const int c = blockIdx.x;           // 0..111 (>=100 are zero pad rows)
    const int tid = threadIdx.x;
    __shared__ float r1[256], r2[256];
    float lsum = 0.f, lsq = 0.f;

    if (c < N_WAY) {
        int idx[K_SHOT];
#pragma unroll
        for (int k = 0; k < K_SHOT; ++k) idx[k] = order[c * K_SHOT + k];
        for (int d = tid; d < D_EMB; d += 256) {
            float v[K_SHOT];
            float sum = 0.f;
#pragma unroll
            for (int k = 0; k < K_SHOT; ++k) {
                v[k] = zs[(size_t)idx[k] * D_EMB + d];
                sum += v[k];
            }
            // insertion sort (ascending), lower median = v[4]
#pragma unroll
            for (int i = 1; i < K_SHOT; ++i) {
                float key = v[i]; int j = i - 1;
                while (j >= 0 && v[j] > key) { v[j + 1] = v[j]; --j; }
                v[j + 1] = key;
            }
            const float t = 0.5f * (v[(K_SHOT - 1) / 2] + sum * (1.0f / K_SHOT));
            ztot[(size_t)c * D_EMB + d] = t;
            lsum += t; lsq += t * t;
        }
    } else {
        for (int d = tid; d < D_EMB; d += 256) ztot[(size_t)c * D_EMB + d] = 0.f;
    }

    r1[tid] = lsum; r2[tid] = lsq;
    __syncthreads();
    for (int s = 128; s > 0; s >>= 1) {
        if (tid < s) { r1[tid] += r1[tid + s]; r2[tid] += r2[tid + s]; }
        __syncthreads();
    }
    if (tid == 0) { ps[c] = r1[0]; p2[c] = r2[0]; }
}

// ---------------- per-query stats: ||q||^2 and sum(q) ----------------
__global__ __launch_bounds__(256)
void qstats_kernel(const float* __restrict__ zq, float* __restrict__ q2, float* __restrict__ qs) {
    const int wave = threadIdx.x >> 5, lane = threadIdx.x & 31;
    const int row = blockIdx.x * 8 + wave;
    const float* p = zq + (size_t)row * D_EMB;
    float s = 0.f, s2 = 0.f;
#pragma unroll
    for (int t = 0; t < D_EMB / 32; ++t) {
        const float v = p[lane + 32 * t];
        s += v; s2 += v * v;
    }
#pragma unroll
    for (int m = 16; m >= 1; m >>= 1) {
        s  += __shfl_xor(s, m, 32);
        s2 += __shfl_xor(s2, m, 32);
    }
    if (lane == 0) { qs[row] = s; q2[row] = s2; }
}

// ---------------- distance GEMM + epilogue: out[8192 x 100] ----------------
#define PSTRIDE_B 224  // 2*112, already 32 mod 64 -> disjoint bank halves, no pad needed
#define KTILES_D (D_EMB / 32)

__global__ __launch_bounds__(256)
void dist_kernel(const float* __restrict__ zq, const float* __restrict__ zt,
                 const float* __restrict__ q2, const float* __restrict__ qs,
                 const float* __restrict__ p2, const float* __restrict__ ps,
                 float* __restrict__ out) {
    __shared__ float As[2][16][PSTRIDE];
    __shared__ float Bs[2][16][PSTRIDE_B];

    const int tid  = threadIdx.x;
    const int row0 = blockIdx.x * 128;
    const int wave = tid >> 5, lane = tid & 31;
    const int lm = lane & 15, hi = lane >> 4;
    const int mw = wave * 16;                 // 8 waves * 16 rows = 128 rows

    const int arow = tid >> 3;
    const int akc  = (tid & 7) * 4;
    const int akp  = akc >> 1;

    float4 ra[4], rbv[4];

    auto load_global = [&](int kt) {
        const int k0 = kt * 32;
#pragma unroll
        for (int r = 0; r < 4; ++r)
            ra[r] = *(const float4*)(zq + (size_t)(row0 + arow + 32 * r) * D_EMB + k0 + akc);
#pragma unroll
        for (int t = 0; t < 4; ++t) {
            const int idx = tid + 256 * t;
            if (idx < N_PROTO_PAD * 8)
                rbv[t] = *(const float4*)(zt + (size_t)(idx >> 3) * D_EMB + k0 + (idx & 7) * 4);
        }
    };
    auto store_lds = [&](int buf) {
#pragma unroll
        for (int r = 0; r < 4; ++r) {
            const int row = arow + 32 * r;
            *(float2*)&As[buf][akp    ][2 * row] = make_float2(ra[r].x, ra[r].y);
            *(float2*)&As[buf][akp + 1][2 * row] = make_float2(ra[r].z, ra[r].w);
        }
#pragma unroll
        for (int t = 0; t < 4; ++t) {
            const int idx = tid + 256 * t;
            if (idx < N_PROTO_PAD * 8) {
                const int n  = idx >> 3;
                const int kp = (idx & 7) * 2;
                *(float2*)&Bs[buf][kp    ][2 * n] = make_float2(rbv[t].x, rbv[t].y);
                *(float2*)&Bs[buf][kp + 1][2 * n] = make_float2(rbv[t].z, rbv[t].w);
            }
        }
    };

    v8f acc[7];
#pragma unroll
    for (int j = 0; j < 7; ++j) acc[j] = (v8f){0.f,0.f,0.f,0.f,0.f,0.f,0.f,0.f};

    load_global(0);
    store_lds(0);
    int cur = 0;

    for (int kt = 0; kt < KTILES_D; ++kt) {
        __syncthreads();
        if (kt + 1 < KTILES_D) load_global(kt + 1);

        const float (*pA)[PSTRIDE]   = As[cur];
        const float (*pB)[PSTRIDE_B] = Bs[cur];
#pragma unroll
        for (int ks = 0; ks < 8; ++ks) {
            const int kp = ks * 2 + hi;
            const v2f a = *(const v2f*)&pA[kp][2 * (mw + lm)];
#pragma unroll
            for (int j = 0; j < 7; ++j) {
                const v2f b = *(const v2f*)&pB[kp][2 * (16 * j + lm)];
                acc[j] = __builtin_amdgcn_wmma_f32_16x16x4_f32(
                    false, a, false, b, (short)0, acc[j], false, false);
            }
        }

        if (kt + 1 < KTILES_D) store_lds(cur ^ 1);
        cur ^= 1;
    }

    // epilogue: -sqrt(max(q2 + p2 - 2*cross + 2*eps*(qs - ps) + D*eps^2, 0))
    const float cD = (float)D_EMB * EPSV * EPSV;
#pragma unroll
    for (int j = 0; j < 7; ++j) {
        const int col = 16 * j + lm;
        if (col < N_WAY) {
            const float pp2 = p2[col], pps = ps[col];
#pragma unroll
            for (int e = 0; e < 8; ++e) {
                const int row = row0 + mw + 8 * hi + e;
                const float sq = q2[row] + pp2 - 2.0f * acc[j][e]
                               + 2.0f * EPSV * (qs[row] - pps) + cD;
                out[(size_t)row * N_WAY + col] = -sqrtf(fmaxf(sq, 0.0f));
            }
        }
    }
}

// ---------------- host launch ----------------
extern "C" void kernel_launch(void* const* d_in, const int* in_sizes, int n_in,
                              void* d_out, int out_size, void* d_ws, size_t ws_size,
                              hipStream_t stream) {
    const float* support = (const float*)d_in[0];
    const int*   labels  = (const int*)d_in[1];
    const float* query   = (const float*)d_in[2];
    const float* Wm      = (const float*)d_in[3];
    const float* bias    = (const float*)d_in[4];
    float* out = (float*)d_out;

    float* ws = (float*)d_ws;
    float* zq = ws + OFF_ZQ;
    float* zs = ws + OFF_ZS;
    float* zt = ws + OFF_ZT;
    float* p2 = ws + OFF_P2;
    float* ps = ws + OFF_PS;
    float* q2 = ws + OFF_Q2;
    float* qs = ws + OFF_QS;
    int*   order = (int*)(ws + OFF_ORDER);

    build_order_kernel<<<1, 128, 0, stream>>>(labels, order);

    // z_query = Q @ W + b   (64 x 8 tiles of 128x128)
    gemm_bias_kernel<<<dim3(N_QUERY / 128, D_EMB / 128), 256, 0, stream>>>(query, Wm, bias, zq, N_QUERY);
    // z_support = S @ W + b (8 x 8 tiles, rows clamped/guarded at 1000)
    gemm_bias_kernel<<<dim3((N_SUPPORT + 127) / 128, D_EMB / 128), 256, 0, stream>>>(support, Wm, bias, zs, N_SUPPORT);

    proto_kernel<<<N_PROTO_PAD, 256, 0, stream>>>(zs, order, zt, p2, ps);
    qstats_kernel<<<N_QUERY / 8, 256, 0, stream>>>(zq, q2, qs);
    dist_kernel<<<N_QUERY / 128, 256, 0, stream>>>(zq, zt, q2, qs, p2, ps, out);
}